// StockFormer_48730698940774
// MI455X (gfx1250) — compile-verified
//
#include <hip/hip_runtime.h>
#include <stdint.h>

// Haar wavelet decoupling, bandwidth-bound streaming kernel for gfx1250.
//   x: [B=8, S=512, N=100, F=6] fp32, inner row = N*F = 600 floats (150 float4)
//   out = concat(X_l flat, X_h flat)
//
// Data path: CDNA5 async global->LDS copies (ASYNCcnt) -> LDS read -> VALU
// butterfly -> 128-bit global stores.

#define INV_SQRT2F 0.70710678118654752440f
#define INNER4 150   // float4 elements per time row (600 floats / 4)

__global__ __launch_bounds__(256) void haar_decouple_kernel(
    const float* __restrict__ x, float* __restrict__ out,
    int nItems,     // total float4 work items = pairs * INNER4
    int outElems)   // floats per output tensor (X_l size == X_h size)
{
    __shared__ float4 sE[256];
    __shared__ float4 sO[256];

    const int tid = threadIdx.x;
    const int w   = blockIdx.x * 256 + tid;
    const bool active = (w < nItems);

    // pair index and float4-column within the pair's row
    const int pair = w / INNER4;
    const int col  = w - pair * INNER4;
    const long eIdx = (long)pair * (2 * INNER4) + col;  // even row float4 index
    const long oIdx = eIdx + INNER4;                    // odd  row float4 index

    const float4* x4 = (const float4*)x;

    if (active) {
        // LDS byte offsets of this thread's private staging slots.
        // Low 32 bits of a shared-aperture flat address == LDS offset.
        unsigned ldsE = (unsigned)(uintptr_t)(&sE[tid]);
        unsigned ldsO = (unsigned)(uintptr_t)(&sO[tid]);
        unsigned long long ge = (unsigned long long)(uintptr_t)(x4 + eIdx);
        unsigned long long go = (unsigned long long)(uintptr_t)(x4 + oIdx);

        // CDNA5 async copy: global memory -> LDS, tracked with ASYNCcnt.
        asm volatile("global_load_async_to_lds_b128 %0, %1, off"
                     :: "v"(ldsE), "v"(ge) : "memory");
        asm volatile("global_load_async_to_lds_b128 %0, %1, off"
                     :: "v"(ldsO), "v"(go) : "memory");
    }

    // Each thread consumes only its own slots; per-wave ASYNCcnt wait is
    // sufficient, no workgroup barrier required.
    asm volatile("s_wait_asynccnt 0" ::: "memory");

    if (active) {
        float4 e = sE[tid];
        float4 o = sO[tid];

        // Match reference rounding exactly: two sequential fp32 multiplies.
        float4 a, d, dn;
        a.x = ((e.x + o.x) * INV_SQRT2F) * INV_SQRT2F;
        a.y = ((e.y + o.y) * INV_SQRT2F) * INV_SQRT2F;
        a.z = ((e.z + o.z) * INV_SQRT2F) * INV_SQRT2F;
        a.w = ((e.w + o.w) * INV_SQRT2F) * INV_SQRT2F;

        d.x = ((e.x - o.x) * INV_SQRT2F) * INV_SQRT2F;
        d.y = ((e.y - o.y) * INV_SQRT2F) * INV_SQRT2F;
        d.z = ((e.z - o.z) * INV_SQRT2F) * INV_SQRT2F;
        d.w = ((e.w - o.w) * INV_SQRT2F) * INV_SQRT2F;

        dn.x = -d.x; dn.y = -d.y; dn.z = -d.z; dn.w = -d.w;

        float4* Xl = (float4*)out;
        float4* Xh = (float4*)(out + outElems);

        Xl[eIdx] = a;   // X_l even
        Xl[oIdx] = a;   // X_l odd (same value)
        Xh[eIdx] = d;   // X_h even
        Xh[oIdx] = dn;  // X_h odd (negated)
    }
}

extern "C" void kernel_launch(void* const* d_in, const int* in_sizes, int n_in,
                              void* d_out, int out_size, void* d_ws, size_t ws_size,
                              hipStream_t stream) {
    (void)n_in; (void)d_ws; (void)ws_size;

    const float* x = (const float*)d_in[0];
    float* out = (float*)d_out;

    const int total    = in_sizes[0];   // B*S*N*F = 2,457,600
    const int outElems = total;         // each of X_l / X_h
    const int nItems   = total / 8;     // (pairs) * (INNER/4) = total / (2*4)
    const int blocks   = (nItems + 255) / 256;   // 1200 for the reference shape

    haar_decouple_kernel<<<blocks, 256, 0, stream>>>(x, out, nItems, outElems);
    (void)out_size;
}